// GNNEncoder_42640435315163
// MI455X (gfx1250) — compile-verified
//
#include <hip/hip_runtime.h>

typedef float v2f __attribute__((ext_vector_type(2)));
typedef float v8f __attribute__((ext_vector_type(8)));

#define HIDC 64
#define EPSV 1e-5f

// ---------------- degree / normalization ----------------

__global__ void deg_init_k(float* deg, int n) {
  int i = blockIdx.x * blockDim.x + threadIdx.x;
  if (i < n) deg[i] = 1.0f;                      // self-loop weight
}

__global__ void deg_acc_k(const int* __restrict__ col, const float* __restrict__ w,
                          float* deg, int e) {
  int i = blockIdx.x * blockDim.x + threadIdx.x;
  if (i < e) atomicAdd(&deg[col[i]], w[i]);
}

__global__ void make_dis_k(float* deg_dis, float* selfn, int n) {
  int i = blockIdx.x * blockDim.x + threadIdx.x;
  if (i < n) {
    float d = deg_dis[i];
    float r = d > 0.0f ? rsqrtf(d) : 0.0f;
    deg_dis[i] = r;
    selfn[i] = r * r;                            // dis[i]*1.0*dis[i]
  }
}

__global__ void make_norm_k(const int* __restrict__ row, const int* __restrict__ col,
                            const float* __restrict__ w, const float* __restrict__ dis,
                            float* norm, int e) {
  int i = blockIdx.x * blockDim.x + threadIdx.x;
  if (i < e) norm[i] = dis[row[i]] * w[i] * dis[col[i]];
}

// ---------------- GEMM: T[N x 64] = A[N x K] @ W[K x 64] (fp32 WMMA) -------
// One 16x16 output tile per wave; 8 waves/block -> 32 rows x 64 cols per block.

__global__ __launch_bounds__(256) void gemm_wmma_k(const float* __restrict__ A,
                                                   const float* __restrict__ W,
                                                   float* __restrict__ T,
                                                   int n, int K) {
  const int lane = threadIdx.x & 31;
  const int wave = threadIdx.x >> 5;
  const int tile_m = blockIdx.x * 32 + (wave >> 2) * 16;
  const int tile_n = (wave & 3) * 16;
  const int l15 = lane & 15;
  const int hi  = lane >> 4;                     // 0: K+{0,1}  1: K+{2,3}
  const int kboff = hi * 2;

  int arow = tile_m + l15;
  if (arow >= n) arow = n - 1;                   // safe clamp (no EXEC divergence)
  const float* Arow = A + (size_t)arow * K;
  const int bcol = tile_n + l15;

  v8f acc = {0.f, 0.f, 0.f, 0.f, 0.f, 0.f, 0.f, 0.f};
  for (int kk = 0; kk < K; kk += 4) {
    const int kb = kk + kboff;
    v2f a = *(const v2f*)(Arow + kb);            // A[M=lane%16][kb], [kb+1]
    v2f b;
    b.x = W[(size_t)kb * HIDC + bcol];           // B[kb][N=lane%16]
    b.y = W[(size_t)(kb + 1) * HIDC + bcol];
    acc = __builtin_amdgcn_wmma_f32_16x16x4_f32(false, a, false, b,
                                                (short)0, acc, false, false);
  }

  const int mbase = tile_m + hi * 8;             // C/D: VGPR r -> M = r + 8*hi
#pragma unroll
  for (int r = 0; r < 8; ++r) {
    int m = mbase + r;
    if (m < n) T[(size_t)m * HIDC + tile_n + l15] = acc[r];
  }
}

// ---------------- scatter: agg = b + selfnorm*t + sum_e norm[e]*t[row[e]] ---

__global__ void scatter_init_k(const float* __restrict__ t,
                               const float* __restrict__ selfn,
                               const float* __restrict__ bias,
                               float* agg, int n) {
  long long idx = (long long)blockIdx.x * blockDim.x + threadIdx.x;
  if (idx < (long long)n * HIDC) {
    int i = (int)(idx >> 6);
    int c = (int)idx & (HIDC - 1);
    agg[idx] = bias[c] + selfn[i] * t[idx];
  }
}

__global__ void scatter_edges_k(const float* __restrict__ t,
                                const float* __restrict__ norm,
                                const int* __restrict__ row,
                                const int* __restrict__ col,
                                float* agg, int e) {
  long long idx = (long long)blockIdx.x * blockDim.x + threadIdx.x;
  int eid = (int)(idx >> 4);                     // 16 threads per edge
  if (eid >= e) return;
  int c4 = ((int)idx & 15) * 4;
  float nv = norm[eid];
  int r = row[eid], cl = col[eid];
  const float4 tv = *(const float4*)(t + (size_t)r * HIDC + c4);
  float* dst = agg + (size_t)cl * HIDC + c4;     // 25.6MB dest -> lives in L2
  atomicAdd(dst + 0, nv * tv.x);
  atomicAdd(dst + 1, nv * tv.y);
  atomicAdd(dst + 2, nv * tv.z);
  atomicAdd(dst + 3, nv * tv.w);
}

// ---------------- BatchNorm (training-mode batch stats) + ReLU --------------

__global__ void zero_stat_k(float* s) {
  if (threadIdx.x < 2 * HIDC) s[threadIdx.x] = 0.0f;
}

__global__ __launch_bounds__(256) void bn_stats_k(const float* __restrict__ v,
                                                  float* stat, int n) {
  __shared__ float ls[4][HIDC];
  __shared__ float lq[4][HIDC];
  int c = threadIdx.x & (HIDC - 1);
  int g = threadIdx.x >> 6;                      // 4 row-groups per block
  float s = 0.0f, q = 0.0f;
  for (int r = blockIdx.x * 4 + g; r < n; r += gridDim.x * 4) {
    float xv = v[(size_t)r * HIDC + c];
    s += xv; q += xv * xv;
  }
  ls[g][c] = s; lq[g][c] = q;
  __syncthreads();
  if (g == 0) {
    s = ls[0][c] + ls[1][c] + ls[2][c] + ls[3][c];
    q = lq[0][c] + lq[1][c] + lq[2][c] + lq[3][c];
    atomicAdd(&stat[c], s);
    atomicAdd(&stat[HIDC + c], q);
  }
}

__global__ void bn_apply_relu_k(const float* __restrict__ v,
                                const float* __restrict__ stat,
                                const float* __restrict__ gm,
                                const float* __restrict__ bt,
                                float* out, int n) {
  long long idx = (long long)blockIdx.x * blockDim.x + threadIdx.x;
  if (idx >= (long long)n * HIDC) return;
  int c = (int)idx & (HIDC - 1);
  float invN = 1.0f / (float)n;
  float mu  = stat[c] * invN;
  float var = stat[HIDC + c] * invN - mu * mu;   // biased variance
  float x = (v[idx] - mu) * rsqrtf(var + EPSV) * gm[c] + bt[c];
  out[idx] = x > 0.0f ? x : 0.0f;
}

__global__ void fill_mask_k(float* m, int n) {
  int i = blockIdx.x * blockDim.x + threadIdx.x;
  if (i < n) m[i] = 1.0f;
}

// ---------------- launch ----------------------------------------------------

static inline unsigned dup(long long a, int b) { return (unsigned)((a + b - 1) / b); }

extern "C" void kernel_launch(void* const* d_in, const int* in_sizes, int n_in,
                              void* d_out, int out_size, void* d_ws, size_t ws_size,
                              hipStream_t stream) {
  const float* x   = (const float*)d_in[0];
  const float* ew  = (const float*)d_in[1];
  const float* Ws[3]  = {(const float*)d_in[2],  (const float*)d_in[6],  (const float*)d_in[10]};
  const float* bs[3]  = {(const float*)d_in[3],  (const float*)d_in[7],  (const float*)d_in[11]};
  const float* gs[3]  = {(const float*)d_in[4],  (const float*)d_in[8],  (const float*)d_in[12]};
  const float* bes[3] = {(const float*)d_in[5],  (const float*)d_in[9],  (const float*)d_in[13]};
  const int* ei = (const int*)d_in[14];

  const int E   = in_sizes[1];
  const int hid = in_sizes[3];                   // 64
  const int INC = in_sizes[2] / hid;             // 128
  const int N   = in_sizes[0] / INC;             // 100000
  const int* row = ei;
  const int* col = ei + E;

  float* ws    = (float*)d_ws;
  float* norm  = ws;                             // E
  float* dis   = norm + E;                       // N (deg, then dis in place)
  float* selfn = dis + N;                        // N
  float* t     = selfn + N;                      // N*64
  float* agg   = t + (size_t)N * HIDC;           // N*64
  float* stat  = agg + (size_t)N * HIDC;         // 128

  const int B = 256;

  deg_init_k<<<dup(N, B), B, 0, stream>>>(dis, N);
  deg_acc_k<<<dup(E, B), B, 0, stream>>>(col, ew, dis, E);
  make_dis_k<<<dup(N, B), B, 0, stream>>>(dis, selfn, N);
  make_norm_k<<<dup(E, B), B, 0, stream>>>(row, col, ew, dis, norm, E);

  const float* layer_in[3] = {x, agg, agg};
  const int    layer_K[3]  = {INC, HIDC, HIDC};
  float*       layer_out[3] = {agg, agg, (float*)d_out};

  for (int l = 0; l < 3; ++l) {
    gemm_wmma_k<<<dup(N, 32), B, 0, stream>>>(layer_in[l], Ws[l], t, N, layer_K[l]);
    scatter_init_k<<<dup((long long)N * HIDC, B), B, 0, stream>>>(t, selfn, bs[l], agg, N);
    scatter_edges_k<<<dup((long long)E * 16, B), B, 0, stream>>>(t, norm, row, col, agg, E);
    zero_stat_k<<<1, 128, 0, stream>>>(stat);
    bn_stats_k<<<512, B, 0, stream>>>(agg, stat, N);
    bn_apply_relu_k<<<dup((long long)N * HIDC, B), B, 0, stream>>>(agg, stat, gs[l], bes[l],
                                                                  layer_out[l], N);
  }

  fill_mask_k<<<dup(N, B), B, 0, stream>>>((float*)d_out + (size_t)N * HIDC, N);
}